// MaxUnpooling2D_80900003987460
// MI455X (gfx1250) — compile-verified
//
#include <hip/hip_runtime.h>
#include <cstdint>

// Reference shapes (fixed): updates/mask [16,128,128,64], out [16,256,256,64]
// in-elements per batch  = 128*128*64 = 2^20
// out-elements per batch = 256*256*64 = 2^22
static constexpr uint32_t IN_PB_LOG2  = 20;
static constexpr uint32_t OUT_PB_LOG2 = 22;

typedef float     __attribute__((ext_vector_type(4))) f32x4;
typedef long long __attribute__((ext_vector_type(2))) i64x2;

// ---------------------------------------------------------------------------
// Kernel 1: zero the 268 MB output with 128-bit stores (global_store_b128).
// ---------------------------------------------------------------------------
__global__ __launch_bounds__(256) void zero_f32x4_kernel(f32x4* __restrict__ out,
                                                         uint32_t n4) {
  uint32_t i = blockIdx.x * blockDim.x + threadIdx.x;
  if (i < n4) {
    f32x4 z = {0.0f, 0.0f, 0.0f, 0.0f};
    out[i] = z;
  }
}

// ---------------------------------------------------------------------------
// Kernel 2: scatter-add. 4 elements per thread:
//   1x global_load_b128 (updates, NT) + 2x global_load_b128 (mask, NT)
//   4x global_atomic_add_f32 (no-return, SADDR form, device scope)
// ---------------------------------------------------------------------------
__device__ __forceinline__ void atomic_add_f32_noret(unsigned long long sbase,
                                                     uint32_t byte_off,
                                                     float val) {
  // Non-returning f32 atomic add: executes at L2 atomic units, tracked on
  // STOREcnt; no wait needed (implicit S_WAIT_IDLE at S_ENDPGM).
  asm volatile("global_atomic_add_f32 %0, %1, %2 scope:SCOPE_DEV"
               :
               : "v"(byte_off), "v"(val), "s"(sbase)
               : "memory");
}

__global__ __launch_bounds__(256) void scatter_add_kernel(
    const f32x4* __restrict__ updates, const i64x2* __restrict__ mask,
    unsigned long long out_base, uint32_t n4) {
  uint32_t i = blockIdx.x * blockDim.x + threadIdx.x;
  if (i >= n4) return;

  // Streamed-once inputs: non-temporal so they don't evict the scatter
  // target's working set from L2.
  f32x4 u   = __builtin_nontemporal_load(&updates[i]);
  i64x2 m01 = __builtin_nontemporal_load(&mask[2 * i]);
  i64x2 m23 = __builtin_nontemporal_load(&mask[2 * i + 1]);

  // All 4 elements of a quad share a batch (2^20 divisible by 4).
  uint32_t e0        = i << 2;                                  // flat in-index
  uint32_t batch_off = (e0 >> IN_PB_LOG2) << OUT_PB_LOG2;       // out elems

  // Byte offsets; max (15<<22 + 2^22-1)*4 = 268435452 < 2^31 -> 32-bit SADDR form.
  uint32_t o0 = (batch_off + (uint32_t)m01.x) << 2;
  uint32_t o1 = (batch_off + (uint32_t)m01.y) << 2;
  uint32_t o2 = (batch_off + (uint32_t)m23.x) << 2;
  uint32_t o3 = (batch_off + (uint32_t)m23.y) << 2;

  atomic_add_f32_noret(out_base, o0, u.x);
  atomic_add_f32_noret(out_base, o1, u.y);
  atomic_add_f32_noret(out_base, o2, u.z);
  atomic_add_f32_noret(out_base, o3, u.w);
}

extern "C" void kernel_launch(void* const* d_in, const int* in_sizes, int n_in,
                              void* d_out, int out_size, void* d_ws, size_t ws_size,
                              hipStream_t stream) {
  const f32x4* updates = (const f32x4*)d_in[0];        // float32
  const i64x2* mask    = (const i64x2*)d_in[1];        // int64
  float*       out     = (float*)d_out;

  const uint32_t n_in_elems  = (uint32_t)in_sizes[0];  // 16,777,216
  const uint32_t n_out_elems = (uint32_t)out_size;     // 67,108,864

  // 1) Zero the output (harness poisons d_out; must re-zero every call).
  {
    uint32_t n4     = n_out_elems / 4;                 // 16,777,216
    uint32_t blocks = (n4 + 255) / 256;                // 65,536
    zero_f32x4_kernel<<<blocks, 256, 0, stream>>>((f32x4*)out, n4);
  }

  // 2) Scatter-add with hardware f32 atomics.
  {
    uint32_t n4     = n_in_elems / 4;                  // 4,194,304
    uint32_t blocks = (n4 + 255) / 256;                // 16,384
    scatter_add_kernel<<<blocks, 256, 0, stream>>>(
        updates, mask, (unsigned long long)out, n4);
  }
}